// SVDPP_18476949307878
// MI455X (gfx1250) — compile-verified
//
#include <hip/hip_runtime.h>

typedef float v2f __attribute__((ext_vector_type(2)));
typedef float v8f __attribute__((ext_vector_type(8)));

#define EMB 64
#define IPW 256          // items per wave in the segment-sum kernel
#define TILES_PER_BLOCK 4 // 4 waves/block in the WMMA batch kernel

// ---------------------------------------------------------------------------
// Kernel 1: CSR-style segment sum of implicit item factors.
// One wave per contiguous chunk of IPW interactions. Each lane owns 2 of the
// 64 embedding dims (float2). Runs of identical segment_ids are accumulated in
// registers; flush via global f32 atomics only at segment boundaries.
// ---------------------------------------------------------------------------
__global__ void svdpp_segsum(const int* __restrict__ seg_ids,
                             const int* __restrict__ items,
                             const float* __restrict__ iif,
                             float* __restrict__ agg,
                             unsigned int* __restrict__ counts,
                             int T) {
    const int gtid = blockIdx.x * blockDim.x + threadIdx.x;
    const int wave = gtid >> 5;
    const int lane = threadIdx.x & 31;

    long start = (long)wave * IPW;
    if (start >= T) return;
    long end = start + IPW;
    if (end > T) end = T;

    float accx = 0.0f, accy = 0.0f;
    int cur = seg_ids[start];
    unsigned int run = 0;

    for (long t = start; t < end; ++t) {
        int s = seg_ids[t];
        if (s != cur) {
            float* dst = agg + (size_t)cur * EMB + 2 * lane;
            atomicAdd(dst, accx);
            atomicAdd(dst + 1, accy);
            if (lane == 0) atomicAdd(&counts[cur], run);
            accx = 0.0f; accy = 0.0f; run = 0; cur = s;
        }
        int it = items[t];
        const float2 y = *(const float2*)(iif + (size_t)it * EMB + 2 * lane);
        accx += y.x;
        accy += y.y;
        ++run;
    }
    float* dst = agg + (size_t)cur * EMB + 2 * lane;
    atomicAdd(dst, accx);
    atomicAdd(dst + 1, accy);
    if (lane == 0) atomicAdd(&counts[cur], run);
}

// ---------------------------------------------------------------------------
// Kernel 2: batch scoring with V_WMMA_F32_16X16X4_F32.
// One wave handles 16 queries. Stage z = p_u + inv_sqrt*agg_u and q_i into
// LDS, then accumulate D = Z * Q^T over K=64 in 16 chunks of K=4 and read the
// diagonal D[m][m] = dot(z_m, q_m).
//
// Fragment layout (32-bit, wave32):
//   A (16x4):  lanes 0-15 -> M=lane, VGPR0=K0, VGPR1=K1
//              lanes 16-31 -> M=lane-16, VGPR0=K2, VGPR1=K3
//   B (4x16):  VGPR0: lanes 0-15 = row K0, lanes 16-31 = row K2
//              VGPR1: lanes 0-15 = row K1, lanes 16-31 = row K3
//   C/D(16x16): VGPR v: lanes 0-15 -> M=v, lanes 16-31 -> M=v+8
//   Diagonal D[m][m] lives in lanes 0-7 (m=lane, vgpr=lane) and
//   lanes 16-23 (m=lane-8, vgpr=lane-16).
// ---------------------------------------------------------------------------
__global__ void svdpp_batch_wmma(const int* __restrict__ user_idx,
                                 const int* __restrict__ item_idx,
                                 const float* __restrict__ user_factors,
                                 const float* __restrict__ item_factors,
                                 const float* __restrict__ agg,
                                 const unsigned int* __restrict__ counts,
                                 const float* __restrict__ user_bias,
                                 const float* __restrict__ item_bias,
                                 const float* __restrict__ gmean_p,
                                 float* __restrict__ out,
                                 int B) {
    __shared__ __align__(16) float zT[TILES_PER_BLOCK][16][EMB];
    __shared__ __align__(16) float qT[TILES_PER_BLOCK][16][EMB];

    const int lane = threadIdx.x & 31;
    const int wv   = threadIdx.x >> 5;
    const int wave = blockIdx.x * TILES_PER_BLOCK + wv;
    const int b0   = wave * 16;
    if (b0 >= B) return;                     // wave-uniform: EXEC stays all-1s

    const float gmean = gmean_p[0];

    // ---- stage 16 queries into LDS (each lane owns a float2 of D=64) ----
    #pragma unroll 4
    for (int m = 0; m < 16; ++m) {
        int b = b0 + m;
        if (b >= B) b = B - 1;               // clamp (B is a multiple of 16 anyway)
        const int u = user_idx[b];
        const int i = item_idx[b];
        const unsigned int c = counts[u];
        const float inv = (c > 0) ? rsqrtf((float)c) : 0.0f;
        const float2 p = *(const float2*)(user_factors + (size_t)u * EMB + 2 * lane);
        const float2 a = *(const float2*)(agg          + (size_t)u * EMB + 2 * lane);
        const float2 q = *(const float2*)(item_factors + (size_t)i * EMB + 2 * lane);
        zT[wv][m][2 * lane]     = p.x + inv * a.x;
        zT[wv][m][2 * lane + 1] = p.y + inv * a.y;
        qT[wv][m][2 * lane]     = q.x;
        qT[wv][m][2 * lane + 1] = q.y;
    }

    // ---- WMMA accumulation over K = 64 (16 chunks of 4) ----
    const int half = lane >> 4;              // 0: lanes 0-15, 1: lanes 16-31
    const int l15  = lane & 15;
    v8f c = {};
    #pragma unroll
    for (int kc = 0; kc < 16; ++kc) {
        const int kb = kc * 4 + 2 * half;    // K offset this lane-half supplies
        v2f afrag = *(const v2f*)&zT[wv][l15][kb];
        v2f bfrag = *(const v2f*)&qT[wv][l15][kb];
        c = __builtin_amdgcn_wmma_f32_16x16x4_f32(
                /*neg_a=*/false, afrag,
                /*neg_b=*/false, bfrag,
                /*c_mod=*/(short)0, c,
                /*reuse_a=*/false, /*reuse_b=*/false);
    }

    // ---- extract diagonal and write result ----
    const int k = lane & 7;                  // VGPR index holding this lane's diag
    float dot = c[0];
    #pragma unroll
    for (int j = 1; j < 8; ++j) dot = (k == j) ? c[j] : dot;

    const bool valid = ((lane & 8) == 0);    // lanes 0-7 and 16-23 hold diagonal
    if (valid) {
        const int m = (lane & 7) | (half << 3);
        const int b = b0 + m;
        if (b < B) {
            const int u = user_idx[b];
            const int i = item_idx[b];
            out[b] = gmean + user_bias[u] + item_bias[i] + dot;
        }
    }
}

// ---------------------------------------------------------------------------
extern "C" void kernel_launch(void* const* d_in, const int* in_sizes, int n_in,
                              void* d_out, int out_size, void* d_ws, size_t ws_size,
                              hipStream_t stream) {
    const int*   user_idx     = (const int*)d_in[0];
    const int*   item_idx     = (const int*)d_in[1];
    const int*   flat_items   = (const int*)d_in[2];
    const int*   segment_ids  = (const int*)d_in[3];
    const float* user_factors = (const float*)d_in[4];
    const float* item_factors = (const float*)d_in[5];
    const float* iif          = (const float*)d_in[6];
    const float* user_bias    = (const float*)d_in[7];
    const float* item_bias    = (const float*)d_in[8];
    const float* gmean        = (const float*)d_in[9];

    const int B = in_sizes[0];
    const int T = in_sizes[2];
    const int U = in_sizes[7];

    // workspace: agg[U][EMB] f32, then counts[U] u32
    float*        agg    = (float*)d_ws;
    unsigned int* counts = (unsigned int*)((char*)d_ws + (size_t)U * EMB * sizeof(float));
    const size_t zero_bytes = (size_t)U * EMB * sizeof(float) + (size_t)U * sizeof(unsigned int);
    hipMemsetAsync(d_ws, 0, zero_bytes, stream);

    // Kernel 1: segment sum. 8 waves / block.
    {
        const int waves  = (T + IPW - 1) / IPW;
        const int block  = 256;
        const int wpb    = block / 32;
        const int grid   = (waves + wpb - 1) / wpb;
        svdpp_segsum<<<grid, block, 0, stream>>>(segment_ids, flat_items, iif,
                                                 agg, counts, T);
    }

    // Kernel 2: WMMA batch scoring. 16 queries per wave, 4 waves / block.
    {
        const int tiles = (B + 15) / 16;
        const int grid  = (tiles + TILES_PER_BLOCK - 1) / TILES_PER_BLOCK;
        svdpp_batch_wmma<<<grid, TILES_PER_BLOCK * 32, 0, stream>>>(
            user_idx, item_idx, user_factors, item_factors, agg, counts,
            user_bias, item_bias, gmean, (float*)d_out, B);
    }
}